// SpatioTemporalGNNLSTM_59519656788248
// MI455X (gfx1250) — compile-verified
//
#include <hip/hip_runtime.h>
#include <hip/hip_bf16.h>

typedef __attribute__((ext_vector_type(2))) float v2f;
typedef __attribute__((ext_vector_type(8))) float v8f;

#define B_   16
#define T_   200
#define A_   22
#define FIN_ 8
#define EMB_ 64
#define H_   64
#define LH_  128
#define G4_  512   // 4*LH
#define BT_  3200  // B*T

__device__ __forceinline__ v8f wmma4(v2f a, v2f b, v8f c) {
  // D = A(16x4) * B(4x16) + C, full f32
  return __builtin_amdgcn_wmma_f32_16x16x4_f32(false, a, false, b, (short)0, c, false, false);
}

__device__ __forceinline__ float sigm(float x) { return 1.0f / (1.0f + __expf(-x)); }

// ---------------------------------------------------------------------------
// Kernel 1: embed + ReLU + per-graph mean (GCN on complete graph collapses to
// per-graph mean) + 2 dense GCN layers. One block = 16 graphs (one WMMA
// M-tile). 4 waves, each owns one 16-col N-tile.
// ---------------------------------------------------------------------------
__global__ __launch_bounds__(128) void k_embed_gcn(
    const float* __restrict__ feat, const float* __restrict__ Wemb,
    const float* __restrict__ bemb, const float* __restrict__ Wg1,
    const float* __restrict__ bg1, const float* __restrict__ Wg2,
    const float* __restrict__ bg2, float* __restrict__ gout)
{
  __shared__ float sX[16][65];   // xbar, padded (65 % 64 != 0)
  __shared__ float sH[16][65];   // h1
  const int tid  = threadIdx.x;
  const int lane = tid & 31;
  const int wv   = tid >> 5;           // 0..3
  const int gbase = blockIdx.x * 16;

  // ---- phase a: xbar = mean_a relu(feat @ Wemb + bemb), 1024 outputs ----
  #pragma unroll
  for (int p = 0; p < 8; ++p) {
    int o   = tid + p * 128;
    int gi  = o >> 6;
    int col = o & 63;
    const float* fb = feat + (size_t)(gbase + gi) * (A_ * FIN_);
    float acc = 0.0f;
    for (int a = 0; a < A_; ++a) {
      float s = bemb[col];
      #pragma unroll
      for (int k = 0; k < FIN_; ++k) s += fb[a * FIN_ + k] * Wemb[k * EMB_ + col];
      acc += fmaxf(s, 0.0f);
    }
    sX[gi][col] = acc * (1.0f / (float)A_);
  }
  __syncthreads();

  const int m  = lane & 15;
  const int hi = lane >> 4;
  const int n  = (wv << 4) + m;        // output column 0..63

  // ---- phase b: h1 = relu(xbar @ Wg1 + bg1)  (M=16,N=64,K=64) ----
  v8f acc1;
  #pragma unroll
  for (int r = 0; r < 8; ++r) acc1[r] = bg1[n];
  #pragma unroll
  for (int ks = 0; ks < 16; ++ks) {
    const int kk = 4 * ks + 2 * hi;
    v2f a; a.x = sX[m][kk];            a.y = sX[m][kk + 1];
    v2f b; b.x = Wg1[kk * H_ + n];     b.y = Wg1[(kk + 1) * H_ + n];
    acc1 = wmma4(a, b, acc1);
  }
  #pragma unroll
  for (int r = 0; r < 8; ++r) sH[r + 8 * hi][n] = fmaxf(acc1[r], 0.0f);
  __syncthreads();

  // ---- phase c: g = relu(h1 @ Wg2 + bg2) ----
  v8f acc2;
  #pragma unroll
  for (int r = 0; r < 8; ++r) acc2[r] = bg2[n];
  #pragma unroll
  for (int ks = 0; ks < 16; ++ks) {
    const int kk = 4 * ks + 2 * hi;
    v2f a; a.x = sH[m][kk];            a.y = sH[m][kk + 1];
    v2f b; b.x = Wg2[kk * H_ + n];     b.y = Wg2[(kk + 1) * H_ + n];
    acc2 = wmma4(a, b, acc2);
  }
  #pragma unroll
  for (int r = 0; r < 8; ++r)
    gout[(size_t)(gbase + r + 8 * hi) * H_ + n] = fmaxf(acc2[r], 0.0f);
}

// ---------------------------------------------------------------------------
// Kernel 2: gx_f = g @ W_ih_f^T + (b_ih_f + b_hh_f).  [3200,64] x [64,512].
// One wave per 16x16 output tile: 200 M-tiles * 32 N-tiles = 6400 waves.
// ---------------------------------------------------------------------------
__global__ __launch_bounds__(128) void k_gx(
    const float* __restrict__ g, const float* __restrict__ Wih,
    const float* __restrict__ bih, const float* __restrict__ bhh,
    float* __restrict__ gx)
{
  const int lane = threadIdx.x & 31;
  const int gw   = blockIdx.x * 4 + (threadIdx.x >> 5);
  const int mt = gw >> 5, nt = gw & 31;
  const int m0 = mt * 16;
  const int m  = lane & 15;
  const int hi = lane >> 4;
  const int n  = nt * 16 + m;          // output column 0..511

  const float bsum = bih[n] + bhh[n];
  v8f acc;
  #pragma unroll
  for (int r = 0; r < 8; ++r) acc[r] = bsum;

  const float* arow = g + (size_t)(m0 + m) * H_;
  const float* brow = Wih + (size_t)n * H_;   // W_ih_f[n][k] -> B[k][n]
  #pragma unroll
  for (int ks = 0; ks < 16; ++ks) {
    const int kk = 4 * ks + 2 * hi;
    v2f a; a.x = arow[kk]; a.y = arow[kk + 1];
    v2f b; b.x = brow[kk]; b.y = brow[kk + 1];
    acc = wmma4(a, b, acc);
  }
  #pragma unroll
  for (int r = 0; r < 8; ++r)
    gx[(size_t)(m0 + r + 8 * hi) * G4_ + n] = acc[r];
}

// ---------------------------------------------------------------------------
// Kernel 3: forward LSTM recurrence (200 steps) with W_hh_f resident in VGPRs,
// h in LDS; then backward-LSTM first step; then classifier.
// One workgroup, 8 waves. Wave w owns units u in [16w, 16w+16):
// n-tiles {w, w+8, w+16, w+24} = gates i,f,g,o, element-aligned in register.
// K-step outer / gate inner keeps only ONE A-fragment live (2 VGPRs) and
// interleaves 4 independent wmma accumulation chains for XDL ILP.
// ---------------------------------------------------------------------------
__global__ __launch_bounds__(256) void k_lstm(
    const float* __restrict__ g, const float* __restrict__ gx,
    const float* __restrict__ Whh_f, const float* __restrict__ Wih_b,
    const float* __restrict__ bih_b, const float* __restrict__ bhh_b,
    const float* __restrict__ Wcls, const float* __restrict__ bcls,
    float* __restrict__ out)
{
  __shared__ float sh[16][132];   // forward hidden (stride 132: banks distinct)
  __shared__ float shb[16][132];  // backward hidden (first step)
  const int tid  = threadIdx.x;
  const int lane = tid & 31;
  const int w    = tid >> 5;     // 0..7
  const int m    = lane & 15;
  const int hi   = lane >> 4;
  const int u    = w * 16 + m;   // unit index within a 128-wide gate

  // ---- preload W_hh_f fragments into VGPRs (256 regs/wave, reused 200x) ----
  v2f bfrag[4][32];
  #pragma unroll
  for (int gi = 0; gi < 4; ++gi) {
    const int n = (w + 8 * gi) * 16 + m;       // row of W_hh_f (512x128)
    const float* row = Whh_f + (size_t)n * LH_;
    #pragma unroll
    for (int ks = 0; ks < 32; ++ks) {
      const int kk = 4 * ks + 2 * hi;
      v2f v; v.x = row[kk]; v.y = row[kk + 1];
      bfrag[gi][ks] = v;
    }
  }

  float cst[8];
  #pragma unroll
  for (int r = 0; r < 8; ++r) cst[r] = 0.0f;
  for (int i = tid; i < 16 * 132; i += 256) (&sh[0][0])[i] = 0.0f;

  // ---- software-pipelined gate pre-activation loads (gx_f for step t) ----
  float gpre[4][8];
  #pragma unroll
  for (int gi = 0; gi < 4; ++gi) {
    const int n = (w + 8 * gi) * 16 + m;
    const float* gxb = gx + (size_t)(8 * hi) * T_ * G4_ + n;
    #pragma unroll
    for (int r = 0; r < 8; ++r)
      gpre[gi][r] = gxb[((size_t)r * T_ + 0) * G4_];
  }
  __syncthreads();

  // ---- forward recurrence ----
  for (int t = 0; t < T_; ++t) {
    v8f acc[4];
    #pragma unroll
    for (int gi = 0; gi < 4; ++gi)
      #pragma unroll
      for (int r = 0; r < 8; ++r) acc[gi][r] = gpre[gi][r];

    // prefetch step t+1's pre-activations; latency hides under 128 wmma
    if (t + 1 < T_) {
      #pragma unroll
      for (int gi = 0; gi < 4; ++gi) {
        const int n = (w + 8 * gi) * 16 + m;
        const float* gxb = gx + (size_t)(8 * hi) * T_ * G4_ + n;
        #pragma unroll
        for (int r = 0; r < 8; ++r)
          gpre[gi][r] = gxb[((size_t)r * T_ + (t + 1)) * G4_];
      }
    }

    // z += h @ W_hh^T : one A-fragment live, 4 interleaved wmma chains
    #pragma unroll
    for (int ks = 0; ks < 32; ++ks) {
      const int kk = 4 * ks + 2 * hi;
      v2f a; a.x = sh[m][kk]; a.y = sh[m][kk + 1];
      #pragma unroll
      for (int gi = 0; gi < 4; ++gi)
        acc[gi] = wmma4(a, bfrag[gi][ks], acc[gi]);
    }
    __syncthreads();   // all reads of old h done before anyone writes new h

    // gates i,f,g,o (PyTorch order) — element-aligned, no cross-lane traffic
    #pragma unroll
    for (int r = 0; r < 8; ++r) {
      const float c = sigm(acc[1][r]) * cst[r] + sigm(acc[0][r]) * tanhf(acc[2][r]);
      cst[r] = c;
      sh[r + 8 * hi][u] = sigm(acc[3][r]) * tanhf(c);
    }
    __syncthreads();   // new h visible before next step's reads
  }

  // ---- backward LSTM, first step only: input g[:, T-1], zero state ----
  {
    v8f acc[4];
    #pragma unroll
    for (int gi = 0; gi < 4; ++gi) {
      const int n = (w + 8 * gi) * 16 + m;
      const float bs = bih_b[n] + bhh_b[n];
      #pragma unroll
      for (int r = 0; r < 8; ++r) acc[gi][r] = bs;
    }
    const float* grow = g + (size_t)(m * T_ + (T_ - 1)) * H_;
    #pragma unroll
    for (int ks = 0; ks < 16; ++ks) {
      const int kk = 4 * ks + 2 * hi;
      v2f a; a.x = grow[kk]; a.y = grow[kk + 1];
      #pragma unroll
      for (int gi = 0; gi < 4; ++gi) {
        const float* brow = Wih_b + (size_t)((w + 8 * gi) * 16 + m) * H_;
        v2f b; b.x = brow[kk]; b.y = brow[kk + 1];
        acc[gi] = wmma4(a, b, acc[gi]);
      }
    }
    #pragma unroll
    for (int r = 0; r < 8; ++r) {
      const float c = sigm(acc[0][r]) * tanhf(acc[2][r]);   // c_prev = 0
      shb[r + 8 * hi][u] = sigm(acc[3][r]) * tanhf(c);
    }
  }
  __syncthreads();

  // ---- classifier: out[b] = [hf, hb] @ W_cls + b_cls ----
  if (tid < 16) {
    float s = bcls[0];
    for (int j = 0; j < LH_; ++j)
      s += sh[tid][j] * Wcls[j] + shb[tid][j] * Wcls[LH_ + j];
    out[tid] = s;
  }
}

// ---------------------------------------------------------------------------
extern "C" void kernel_launch(void* const* d_in, const int* in_sizes, int n_in,
                              void* d_out, int out_size, void* d_ws, size_t ws_size,
                              hipStream_t stream) {
  const float* feat  = (const float*)d_in[0];
  const float* Wemb  = (const float*)d_in[1];
  const float* bemb  = (const float*)d_in[2];
  const float* Wg1   = (const float*)d_in[3];
  const float* bg1   = (const float*)d_in[4];
  const float* Wg2   = (const float*)d_in[5];
  const float* bg2   = (const float*)d_in[6];
  const float* Wih_f = (const float*)d_in[7];
  const float* Whh_f = (const float*)d_in[8];
  const float* bih_f = (const float*)d_in[9];
  const float* bhh_f = (const float*)d_in[10];
  const float* Wih_b = (const float*)d_in[11];
  // d_in[12] = W_hh_b: unused (h0 = 0 for the single backward step)
  const float* bih_b = (const float*)d_in[13];
  const float* bhh_b = (const float*)d_in[14];
  const float* Wcls  = (const float*)d_in[15];
  const float* bcls  = (const float*)d_in[16];

  float* g  = (float*)d_ws;                    // [3200, 64]
  float* gx = g + (size_t)BT_ * H_;            // [3200, 512]

  k_embed_gcn<<<BT_ / 16, 128, 0, stream>>>(feat, Wemb, bemb, Wg1, bg1, Wg2, bg2, g);
  k_gx<<<(BT_ / 16) * 32 / 4, 128, 0, stream>>>(g, Wih_f, bih_f, bhh_f, gx);
  k_lstm<<<1, 256, 0, stream>>>(g, gx, Whh_f, Wih_b, bih_b, bhh_b, Wcls, bcls,
                                (float*)d_out);
}